// TEAN_9165460209971
// MI455X (gfx1250) — compile-verified
//
#include <hip/hip_runtime.h>
#include <math.h>

#define B_SZ   256
#define D_DIM  1280
#define N_PIX  49
#define K_CW   8
#define EPS_BN 1e-5f
#define KSPLIT 8              // K-split ways per GEMM block (one per wave)

typedef float v2f __attribute__((ext_vector_type(2)));
typedef float v8f __attribute__((ext_vector_type(8)));

// ---- async global->LDS path (gfx1250): guarded so compile never breaks ----
#if defined(__gfx1250__)
#if __has_builtin(__builtin_amdgcn_global_load_async_to_lds_b32) && \
    __has_builtin(__builtin_amdgcn_s_wait_asynccnt)
#define USE_ASYNC_LDS 1
#endif
#endif
#ifndef USE_ASYNC_LDS
#define USE_ASYNC_LDS 0
#endif

typedef __attribute__((address_space(1))) int gbl_i32;   // global (AS1)
typedef __attribute__((address_space(3))) int lds_i32;   // LDS    (AS3)

// ---------------- workspace layout (float offsets) ----------------
#define OFF_P      0                              // [B, D]   mean(img)
#define OFF_SM     (OFF_P     + B_SZ*D_DIM)       // [B, D]   mean(img+diff)
#define OFF_E      (OFF_SM    + B_SZ*D_DIM)       // [B, K*D] encoding residuals
#define OFF_INV    (OFF_E     + B_SZ*K_CW*D_DIM)  // [B]      1/||E row||
#define OFF_X1     (OFF_INV   + B_SZ)             // [B, 64]
#define OFF_X2     (OFF_X1    + B_SZ*64)          // [B, 64]
#define OFF_ANG    (OFF_X2    + B_SZ*64)          // [B, 128]
#define OFF_OUTER  (OFF_ANG   + B_SZ*128)         // [B, 4096] normalized outer
#define OFF_FC     (OFF_OUTER + B_SZ*4096)        // [B, 128]
#define OFF_CC     (OFF_FC    + B_SZ*128)         // [B, 256] concat(enc, ang)

// ---------------- kernel 1: fused pooling reductions (single HBM pass) ----------------
__global__ void pool_reduce_kernel(const float* __restrict__ img,
                                   const float* __restrict__ dif,
                                   float* __restrict__ p,
                                   float* __restrict__ sm) {
  int t = blockIdx.x * blockDim.x + threadIdx.x;      // one (b,d) per thread
  if (t >= B_SZ * D_DIM) return;
  const size_t base = (size_t)t * N_PIX;
  const float* gi = img + base;
  const float* gd = dif + base;
  __builtin_prefetch(gi + N_PIX * 8, 0, 0);           // global_prefetch_b8
  __builtin_prefetch(gd + N_PIX * 8, 0, 0);
  float s1 = 0.f, s2 = 0.f;
#pragma unroll
  for (int i = 0; i < N_PIX; ++i) {
    float vi = gi[i];
    float vd = gd[i];
    s1 += vi;
    s2 += vi + vd;
  }
  const float inv = 1.0f / (float)N_PIX;
  p[t]  = s1 * inv;
  sm[t] = s2 * inv;
}

// ---------------- kernel 2: Deep-TEN encoding, one block per batch ----------------
__global__ __launch_bounds__(512)
void encode_kernel(const float* __restrict__ img,
                   const float* __restrict__ C,
                   const float* __restrict__ S,
                   float* __restrict__ E) {
  __shared__ float s_C[K_CW * D_DIM];                 // 40 KB codeword stage
  __shared__ float s_dot[N_PIX * K_CW];
  __shared__ float s_xsq[N_PIX];
  __shared__ float s_csq[K_CW];
  __shared__ float s_A[N_PIX * K_CW];
  __shared__ float s_asum[K_CW];

  const int t = threadIdx.x;
  const int b = blockIdx.x;
  const float* X = img + (size_t)b * D_DIM * N_PIX;   // [d*49 + n]

  // stage codewords into LDS (async DMA path on gfx1250, ASYNCcnt-tracked)
#if USE_ASYNC_LDS
  for (int i = t; i < K_CW * D_DIM; i += 512) {
    gbl_i32* g = (gbl_i32*)(size_t)(C + i);
    lds_i32* l = (lds_i32*)(unsigned)(size_t)(s_C + i);
    __builtin_amdgcn_global_load_async_to_lds_b32(g, l, 0, 0);
  }
  __builtin_amdgcn_s_wait_asynccnt(0);
  __syncthreads();
#else
  for (int i = t; i < K_CW * D_DIM; i += 512) s_C[i] = C[i];
  __syncthreads();
#endif

  // phase A: dot[n,k], xsq[n], csq[k]
  if (t < N_PIX * K_CW) {
    const int n = t >> 3, k = t & 7;
    const float* Xn = X + n;
    const float* Ck = s_C + k * D_DIM;
    float acc = 0.f;
    for (int d = 0; d < D_DIM; ++d) acc += Xn[(size_t)d * N_PIX] * Ck[d];
    s_dot[t] = acc;
  } else if (t < N_PIX * K_CW + N_PIX) {
    const int n = t - N_PIX * K_CW;
    const float* Xn = X + n;
    float acc = 0.f;
    for (int d = 0; d < D_DIM; ++d) { float x = Xn[(size_t)d * N_PIX]; acc += x * x; }
    s_xsq[n] = acc;
  } else if (t < N_PIX * K_CW + N_PIX + K_CW) {
    const int k = t - (N_PIX * K_CW + N_PIX);
    const float* Ck = s_C + k * D_DIM;
    float acc = 0.f;
    for (int d = 0; d < D_DIM; ++d) { float c = Ck[d]; acc += c * c; }
    s_csq[k] = acc;
  }
  __syncthreads();

  // phase B: softmax over k per pixel
  if (t < N_PIX) {
    float sl[K_CW];
    float mx = -3.402823e38f;
    const float xs = s_xsq[t];
#pragma unroll
    for (int k = 0; k < K_CW; ++k) {
      float v = S[k] * (xs + s_csq[k] - 2.0f * s_dot[t * K_CW + k]);
      sl[k] = v;
      mx = fmaxf(mx, v);
    }
    float sum = 0.f;
#pragma unroll
    for (int k = 0; k < K_CW; ++k) { sl[k] = __expf(sl[k] - mx); sum += sl[k]; }
    const float r = 1.0f / sum;
#pragma unroll
    for (int k = 0; k < K_CW; ++k) s_A[t * K_CW + k] = sl[k] * r;
  }
  __syncthreads();

  if (t < K_CW) {
    float acc = 0.f;
    for (int n = 0; n < N_PIX; ++n) acc += s_A[n * K_CW + t];
    s_asum[t] = acc;
  }
  __syncthreads();

  // phase C: residual aggregation over d
  for (int d = t; d < D_DIM; d += 512) {
    float acc[K_CW];
#pragma unroll
    for (int k = 0; k < K_CW; ++k) acc[k] = 0.f;
    const float* Xd = X + (size_t)d * N_PIX;
    for (int n = 0; n < N_PIX; ++n) {
      const float x = Xd[n];
#pragma unroll
      for (int k = 0; k < K_CW; ++k) acc[k] += s_A[n * K_CW + k] * x;
    }
#pragma unroll
    for (int k = 0; k < K_CW; ++k)
      E[((size_t)b * K_CW + k) * D_DIM + d] = acc[k] - s_asum[k] * s_C[k * D_DIM + d];
  }
}

// ---------------- shared-memory row reduction helper ----------------
__device__ inline float block_reduce_sum(float v, float* sh) {
  const int t = threadIdx.x;
  sh[t] = v;
  __syncthreads();
  for (int s = blockDim.x >> 1; s > 0; s >>= 1) {
    if (t < s) sh[t] += sh[t + s];
    __syncthreads();
  }
  float r = sh[0];
  __syncthreads();
  return r;
}

// ---------------- kernel 3: 1/||E row|| ----------------
__global__ void rownorm_kernel(const float* __restrict__ E, float* __restrict__ inv,
                               int cols) {
  __shared__ float sh[256];
  const int b = blockIdx.x;
  const float* row = E + (size_t)b * cols;
  float acc = 0.f;
  for (int i = threadIdx.x; i < cols; i += 256) { float v = row[i]; acc += v * v; }
  float s = block_reduce_sum(acc, sh);
  if (threadIdx.x == 0) inv[b] = 1.0f / fmaxf(sqrtf(s), 1e-12f);
}

// ---------------- kernel 4: f32 WMMA GEMM, 64x16 super-tile, 8-way K-split ----------------
// Block = 8 waves. Wave w accumulates K-chunk w over a 64x16 tile (4 WMMAs per
// k-step sharing one B fragment), partials reduced across waves in LDS, then a
// scalar epilogue applies rowscale (fused l2-norm), bias, and BatchNorm.
__global__ __launch_bounds__(256)
void gemm_wmma_f32(const float* __restrict__ A, int lda,
                   const float* __restrict__ Bw, int ldb,
                   float* __restrict__ Cm, int ldc,
                   int M, int N, int Kd,
                   const float* __restrict__ rowscale,
                   const float* __restrict__ bias,
                   const float* __restrict__ bn_g,
                   const float* __restrict__ bn_b,
                   const float* __restrict__ bn_m,
                   const float* __restrict__ bn_v) {
  __shared__ float s_red[KSPLIT * 4 * 256];           // 32 KB partial tiles
  const int lane = threadIdx.x & 31;
  const int wave = threadIdx.x >> 5;
  const int ntn  = (N + 15) >> 4;
  const int m0   = (blockIdx.x / ntn) << 6;           // 64-row super-tile
  const int n0   = (blockIdx.x % ntn) << 4;
  const int half = lane >> 4;                         // A: K0/K1 vs K2/K3 halves
  const int l15  = lane & 15;
  const int nn   = n0 + l15;
  const bool nok = nn < N;
  const int ncl  = nok ? nn : 0;

  const int Kc = Kd / KSPLIT;                         // per-wave K chunk (mult of 4)
  const int k0 = wave * Kc;

  const float*  Ar    = A + (size_t)(m0 + l15) * lda + (size_t)k0 + 2 * half;
  const float*  Bc    = Bw + (size_t)(k0 + 2 * half) * ldb + ncl;
  const size_t  mstep = (size_t)16 * lda;

  v8f c0 = {0.f,0.f,0.f,0.f,0.f,0.f,0.f,0.f};
  v8f c1 = c0, c2 = c0, c3 = c0;

  for (int k = 0; k < Kc; k += 4) {
    __builtin_prefetch(Ar + k + 128, 0, 0);
    v2f b;
    float b0 = Bc[(size_t)k * ldb];
    float b1 = Bc[(size_t)k * ldb + ldb];
    b.x = nok ? b0 : 0.0f;                            // v_cndmask; EXEC stays full
    b.y = nok ? b1 : 0.0f;
    v2f a0, a1, a2, a3;
    a0.x = Ar[k];               a0.y = Ar[k + 1];
    a1.x = Ar[mstep + k];       a1.y = Ar[mstep + k + 1];
    a2.x = Ar[2 * mstep + k];   a2.y = Ar[2 * mstep + k + 1];
    a3.x = Ar[3 * mstep + k];   a3.y = Ar[3 * mstep + k + 1];
    c0 = __builtin_amdgcn_wmma_f32_16x16x4_f32(false, a0, false, b, (short)0, c0, false, false);
    c1 = __builtin_amdgcn_wmma_f32_16x16x4_f32(false, a1, false, b, (short)0, c1, false, false);
    c2 = __builtin_amdgcn_wmma_f32_16x16x4_f32(false, a2, false, b, (short)0, c2, false, false);
    c3 = __builtin_amdgcn_wmma_f32_16x16x4_f32(false, a3, false, b, (short)0, c3, false, false);
  }

  // dump partial tiles: s_red[wave][mt][row][col]
  float* dst = s_red + wave * 1024;
#pragma unroll
  for (int r = 0; r < 8; ++r) {
    const int row = r + 8 * half;                     // C layout: VGPR r -> M=r / r+8
    dst[0 * 256 + row * 16 + l15] = c0[r];
    dst[1 * 256 + row * 16 + l15] = c1[r];
    dst[2 * 256 + row * 16 + l15] = c2[r];
    dst[3 * 256 + row * 16 + l15] = c3[r];
  }
  __syncthreads();

  // cross-wave reduce + fused epilogue (1024 outputs / 256 threads)
  for (int o = threadIdx.x; o < 4 * 256; o += 256) {
    float acc = 0.f;
#pragma unroll
    for (int w = 0; w < KSPLIT; ++w) acc += s_red[w * 1024 + o];
    const int mt  = o >> 8;
    const int row = (o >> 4) & 15;
    const int col = o & 15;
    const int mm  = m0 + mt * 16 + row;
    const int cc  = n0 + col;
    if (cc < N) {
      float y = acc;
      if (rowscale) y *= rowscale[mm];
      if (bias)     y += bias[cc];
      if (bn_g)     y = (y - bn_m[cc]) * (bn_g[cc] * rsqrtf(bn_v[cc] + EPS_BN)) + bn_b[cc];
      Cm[(size_t)mm * ldc + cc] = y;
    }
  }
}

// ---------------- kernel 5: normalized bilinear outer product ----------------
__global__ void outer_kernel(const float* __restrict__ x1,
                             const float* __restrict__ x2,
                             float* __restrict__ outer) {
  __shared__ float sh[256];
  const int b = blockIdx.x, t = threadIdx.x;
  const float* r1 = x1 + (size_t)b * 64;
  const float* r2 = x2 + (size_t)b * 64;
  float v1 = (t < 64) ? r1[t] : 0.f;
  float v2 = (t < 64) ? r2[t] : 0.f;
  float s1 = block_reduce_sum(v1 * v1, sh);
  float s2 = block_reduce_sum(v2 * v2, sh);
  const float inv = 1.0f / fmaxf(sqrtf(s1) * sqrtf(s2), 1e-12f);
  float* o = outer + (size_t)b * 4096;
  for (int idx = t; idx < 4096; idx += 256) {
    const int i = idx >> 6, j = idx & 63;
    o[idx] = r2[i] * r1[j] * inv;
  }
}

// ---------------- kernel 6: l2-normalize fc output + concat with ang ----------------
__global__ void finalize_concat_kernel(const float* __restrict__ fco,
                                       const float* __restrict__ ang,
                                       float* __restrict__ cc) {
  __shared__ float sh[128];
  const int b = blockIdx.x, t = threadIdx.x;          // 128 threads
  const float v = fco[(size_t)b * 128 + t];
  sh[t] = v * v;
  __syncthreads();
  for (int s = 64; s > 0; s >>= 1) {
    if (t < s) sh[t] += sh[t + s];
    __syncthreads();
  }
  const float inv = 1.0f / fmaxf(sqrtf(sh[0]), 1e-12f);
  cc[(size_t)b * 256 + t]       = v * inv;
  cc[(size_t)b * 256 + 128 + t] = ang[(size_t)b * 128 + t];
}

// ---------------- host-side GEMM launcher ----------------
static void launch_gemm(const float* A, int lda, const float* Bw, int ldb,
                        float* Cm, int ldc, int M, int N, int K,
                        const float* rowscale, const float* bias,
                        const float* g, const float* be,
                        const float* m, const float* v, hipStream_t s) {
  const int blocks = (M / 64) * ((N + 15) / 16);      // one 64x16 tile per block
  hipLaunchKernelGGL(gemm_wmma_f32, dim3(blocks), dim3(256), 0, s,
                     A, lda, Bw, ldb, Cm, ldc, M, N, K,
                     rowscale, bias, g, be, m, v);
}

extern "C" void kernel_launch(void* const* d_in, const int* in_sizes, int n_in,
                              void* d_out, int out_size, void* d_ws, size_t ws_size,
                              hipStream_t stream) {
  (void)in_sizes; (void)n_in; (void)out_size; (void)ws_size;
  const float* img    = (const float*)d_in[0];
  const float* dif    = (const float*)d_in[1];
  const float* cw     = (const float*)d_in[2];
  const float* scale  = (const float*)d_in[3];
  const float* head_w = (const float*)d_in[4];
  const float* head_b = (const float*)d_in[5];
  const float* head_g = (const float*)d_in[6];
  const float* head_be= (const float*)d_in[7];
  const float* head_m = (const float*)d_in[8];
  const float* head_v = (const float*)d_in[9];
  const float* pool_w = (const float*)d_in[10];
  const float* pool_b = (const float*)d_in[11];
  const float* pool_g = (const float*)d_in[12];
  const float* pool_be= (const float*)d_in[13];
  const float* pool_m = (const float*)d_in[14];
  const float* pool_v = (const float*)d_in[15];
  const float* fc_w   = (const float*)d_in[16];
  const float* fc_b   = (const float*)d_in[17];
  const float* p2_w   = (const float*)d_in[18];
  const float* p2_b   = (const float*)d_in[19];
  const float* p2_g   = (const float*)d_in[20];
  const float* p2_be  = (const float*)d_in[21];
  const float* p2_m   = (const float*)d_in[22];
  const float* p2_v   = (const float*)d_in[23];
  const float* cls_w  = (const float*)d_in[24];
  const float* cls_b  = (const float*)d_in[25];

  float* ws    = (float*)d_ws;
  float* p     = ws + OFF_P;
  float* sm    = ws + OFF_SM;
  float* E     = ws + OFF_E;
  float* invE  = ws + OFF_INV;
  float* x1    = ws + OFF_X1;
  float* x2    = ws + OFF_X2;
  float* ang   = ws + OFF_ANG;
  float* outer = ws + OFF_OUTER;
  float* fco   = ws + OFF_FC;
  float* cc    = ws + OFF_CC;
  float* out   = (float*)d_out;

  // 1) fused pooling: single pass over img + diff_img (128 MB HBM read)
  hipLaunchKernelGGL(pool_reduce_kernel, dim3((B_SZ * D_DIM + 255) / 256),
                     dim3(256), 0, stream, img, dif, p, sm);

  // 2) Deep-TEN encoding per batch (img[b] re-read hits L2; codewords in LDS)
  hipLaunchKernelGGL(encode_kernel, dim3(B_SZ), dim3(512), 0, stream,
                     img, cw, scale, E);

  // 3) row norms of E for the fused l2-normalize
  hipLaunchKernelGGL(rownorm_kernel, dim3(B_SZ), dim3(256), 0, stream,
                     E, invE, K_CW * D_DIM);

  // 4) head: x1 = BN(l2n(E) @ head_w + head_b)   [256,10240]x[10240,64]
  launch_gemm(E, K_CW * D_DIM, head_w, 64, x1, 64, B_SZ, 64, K_CW * D_DIM,
              invE, head_b, head_g, head_be, head_m, head_v, stream);

  // 5) pool: x2 = BN(p @ pool_w + pool_b)        [256,1280]x[1280,64]
  launch_gemm(p, D_DIM, pool_w, 64, x2, 64, B_SZ, 64, D_DIM,
              nullptr, pool_b, pool_g, pool_be, pool_m, pool_v, stream);

  // 6) ang: BN(mean(img+diff) @ pool2_w + b)     [256,1280]x[1280,128]
  launch_gemm(sm, D_DIM, p2_w, 128, ang, 128, B_SZ, 128, D_DIM,
              nullptr, p2_b, p2_g, p2_be, p2_m, p2_v, stream);

  // 7) normalized bilinear outer product (||x2 (x) x1|| = ||x2||*||x1||)
  hipLaunchKernelGGL(outer_kernel, dim3(B_SZ), dim3(256), 0, stream,
                     x1, x2, outer);

  // 8) fc: fco = l2n(outer) @ fc_w + fc_b        [256,4096]x[4096,128]
  launch_gemm(outer, 4096, fc_w, 128, fco, 128, B_SZ, 128, 4096,
              nullptr, fc_b, nullptr, nullptr, nullptr, nullptr, stream);

  // 9) enc = l2n(fco); concat(enc, ang) -> cc [256,256]
  hipLaunchKernelGGL(finalize_concat_kernel, dim3(B_SZ), dim3(128), 0, stream,
                     fco, ang, cc);

  // 10) classifier: out = cc @ cls_w + cls_b     [256,256]x[256,31]
  launch_gemm(cc, 256, cls_w, 31, out, 31, B_SZ, 31, 256,
              nullptr, cls_b, nullptr, nullptr, nullptr, nullptr, stream);
}